// NODE_55430847922304
// MI455X (gfx1250) — compile-verified
//
#include <hip/hip_runtime.h>
#include <hip/hip_bf16.h>
#include <math.h>

// ---------------------------------------------------------------------------
// NODE (Tsit5 over SEIAR with MLP beta(t)) for MI455X / gfx1250.
//
// Phase 1: beta(t) at all 6*8*(T-1) stage times as a batched MLP.
//          Layer-2 64x64 matmul runs on V_WMMA_F32_16X16X4_F32 (full f32).
//          All W_h fragments are preloaded into VGPRs before layer 1 so the
//          softplus transcendentals hide the global-load latency and the 64
//          WMMAs issue back-to-back (one loadcnt wait, not 64).
// Phase 2: single-lane sequential Tsit5 scan (irreducible dependency chain)
//          consuming the precomputed betas from d_ws.
// ---------------------------------------------------------------------------

typedef __attribute__((ext_vector_type(2))) float v2f;
typedef __attribute__((ext_vector_type(8))) float v8f;

// Tsit5 stage offsets c_j (j=0 is the step start).
__device__ __constant__ float c_stage[6] = {
    0.0f, 0.161f, 0.327f, 0.9f, 0.9800255409045097f, 1.0f};

__device__ __forceinline__ float softplus_f(float x) {
  // jax.nn.softplus: max(x,0) + log1p(exp(-|x|))  (overflow-safe)
  return fmaxf(x, 0.0f) + log1pf(__expf(-fabsf(x)));
}

// ---------------------------------------------------------------------------
// Phase 1: one wave per 16 beta-columns. Computes
//   h1 = softplus(w_in * t + b_in)            (64 x 16, built in B layout)
//   h2 = softplus(W_h @ h1 + b_h)             (64 x 16, via 64 f32 WMMAs)
//   beta = sigmoid(1e-4 * (w_out @ h2 + b_out))
// Column n: substep m = n/6, stage j = n%6, interval q = m/8, i = m%8,
//   h = (ts[q+1]-ts[q])/8,  t = ts[q] + i*h + c_j*h   (matches reference assoc)
// ---------------------------------------------------------------------------
__global__ __launch_bounds__(32) void beta_mlp_wmma(
    const float* __restrict__ ts,
    const float* __restrict__ w_in, const float* __restrict__ b_in,
    const float* __restrict__ w_h,  const float* __restrict__ b_h,
    const float* __restrict__ w_out, const float* __restrict__ b_out,
    float* __restrict__ betas, int ncol)
{
  const int lane = threadIdx.x;      // 0..31
  const int cL   = lane & 15;        // column within this wave's 16-col tile
  const int hi   = lane >> 4;        // 0: K rows {4k,4k+1}, 1: {4k+2,4k+3}

  const int col     = blockIdx.x * 16 + cL;
  const bool col_ok = (col < ncol);
  const int  cc     = col_ok ? col : (ncol - 1);     // clamp: keep EXEC all-1s

  // ---- Preload ALL W_h A-fragments first (issue early, consume late). ----
  // A-tile mi,k (16x4): lane L (both halves) -> row 16*mi+L, lanes 0-15 hold
  // K columns {4k,4k+1}, lanes 16-31 hold {4k+2,4k+3}.
  v2f a[4][16];
#pragma unroll
  for (int mi = 0; mi < 4; ++mi) {
    const float* wrow = w_h + (size_t)(16 * mi + cL) * 64 + 2 * hi;
#pragma unroll
    for (int k = 0; k < 16; ++k) {
      a[mi][k] = *(const v2f*)(wrow + 4 * k);   // global_load_b64, clause-able
    }
  }

  // ---- Stage time for this column. ----
  const int m = cc / 6;
  const int j = cc - 6 * m;
  const int q = m >> 3;
  const int i = m & 7;
  const float t0 = ts[q];
  const float hh = (ts[q + 1] - t0) * 0.125f;
  const float t  = t0 + (float)i * hh + c_stage[j] * hh;

  // ---- Layer 1 directly in WMMA B-operand layout (hides A-load latency). --
  // B-tile k (K rows 4k..4k+3, 16 cols): lanes 0-15 hold rows 4k,4k+1 of col
  // cL; lanes 16-31 hold rows 4k+2,4k+3 (mirrors the f32 A 16x4 layout).
  v2f h1[16];
#pragma unroll
  for (int k = 0; k < 16; ++k) {
    const int r0 = 4 * k + 2 * hi;
    h1[k].x = softplus_f(w_in[r0]     * t + b_in[r0]);
    h1[k].y = softplus_f(w_in[r0 + 1] * t + b_in[r0 + 1]);
  }

  // ---- Layer 2: 4 output row-tiles x 16 K-steps of WMMA f32 16x16x4,
  //      then layer-3 partial dot with w_out in the C/D layout. ----
  float partial = 0.0f;
#pragma unroll
  for (int mi = 0; mi < 4; ++mi) {
    v8f acc = {0.f, 0.f, 0.f, 0.f, 0.f, 0.f, 0.f, 0.f};
#pragma unroll
    for (int k = 0; k < 16; ++k) {
      acc = __builtin_amdgcn_wmma_f32_16x16x4_f32(
          /*neg_a=*/false, a[mi][k], /*neg_b=*/false, h1[k],
          /*c_mod=*/(short)0, acc, /*reuse_a=*/false, /*reuse_b=*/false);
    }
    // C layout: VGPR r, lanes 0-15 -> row 16*mi+r, lanes 16-31 -> row 16*mi+8+r
#pragma unroll
    for (int r = 0; r < 8; ++r) {
      const int row = 16 * mi + 8 * hi + r;
      const float h2 = softplus_f(acc[r] + b_h[row]);
      partial = fmaf(w_out[row], h2, partial);
    }
  }

  // Combine the two half-wave partials (rows split across lane halves).
  const float o = partial + __shfl_xor(partial, 16, 32) + b_out[0];
  const float beta = 1.0f / (1.0f + __expf(-1e-4f * o));

  if (hi == 0 && col_ok) betas[col] = beta;
}

// ---------------------------------------------------------------------------
// Phase 2: sequential Tsit5 scan, one lane. 5 independent component chains
// give the scheduler ILP; betas stream from L2 (written by phase 1).
// ---------------------------------------------------------------------------
__global__ __launch_bounds__(32) void tsit5_scan(
    const float* __restrict__ ts,
    const float* __restrict__ state_vec,
    const float* __restrict__ scales,
    const float* __restrict__ betas,
    float* __restrict__ out, int nsub)
{
  if (threadIdx.x != 0 || blockIdx.x != 0) return;

  // Tsit5 tableau (f32-rounded, matching the reference constants)
  const float A21 = 0.161f;
  const float A31 = -0.008480655492356989f, A32 = 0.335480655492357f;
  const float A41 = 2.8971530571054935f, A42 = -6.359448489975075f,
              A43 = 4.3622954328695815f;
  const float A51 = 5.325864828439257f, A52 = -11.748883564062828f,
              A53 = 7.4955393428898365f, A54 = -0.09249506636175525f;
  const float A61 = 5.86145544294642f, A62 = -12.92096931784711f,
              A63 = 8.159367898576159f, A64 = -0.071584973281401f,
              A65 = -0.028269050394068383f;
  const float B1 = 0.09646076681806523f, B2 = 0.01f, B3 = 0.4798896504144996f,
              B4 = 1.379008574103742f, B5 = -3.290069515436081f,
              B6 = 2.324710524099774f;
  // SEIAR constants (EE=0, DD=1, Q=0.5 folded): LL = 0.5*I + A
  const float KK = 0.526f, AAc = 0.244f, IIc = 0.244f;
  const float PKK  = (float)(0.667 * 0.526);          // P*KK
  const float PKKc = (float)((1.0 - 0.667) * 0.526);  // (1-P)*KK
  const float FAA  = (float)(0.98 * 0.244);           // F*AA

  float sc[5], inv[5];
#pragma unroll
  for (int c = 0; c < 5; ++c) { sc[c] = scales[c]; inv[c] = 1.0f / sc[c]; }

  // y0 = softmax(state_vec) / scales ; out row 0 = y0 * scales
  float v[5];
  float mx = state_vec[0];
#pragma unroll
  for (int c = 1; c < 5; ++c) mx = fmaxf(mx, state_vec[c]);
  float sum = 0.0f;
#pragma unroll
  for (int c = 0; c < 5; ++c) { v[c] = expf(state_vec[c] - mx); sum += v[c]; }
  const float rsum = 1.0f / sum;
  float y[5];
#pragma unroll
  for (int c = 0; c < 5; ++c) {
    y[c]   = (v[c] * rsum) * inv[c];
    out[c] = y[c] * sc[c];
  }

  auto rhs = [&](float b, const float* yy, float* k) {
    const float S = yy[0] * sc[0], E = yy[1] * sc[1];
    const float I = yy[2] * sc[2], A = yy[3] * sc[3];
    const float LL = fmaf(0.5f, I, A);
    const float g  = b * S * LL;
    k[0] = (-g) * inv[0];
    k[1] = (g - KK * E) * inv[1];
    k[2] = (PKK * E - AAc * I) * inv[2];
    k[3] = (PKKc * E - IIc * A) * inv[3];
    k[4] = (FAA * I + IIc * A) * inv[4];
  };

  for (int s = 0; s < nsub; ++s) {
    const int q = s >> 3;
    const float h = (ts[q + 1] - ts[q]) * 0.125f;
    const float* bb = betas + 6 * s;
    const float b0 = bb[0], b1 = bb[1], b2 = bb[2],
                b3 = bb[3], b4 = bb[4], b5 = bb[5];

    float k1[5], k2[5], k3[5], k4[5], k5[5], k6[5], yt[5];
    rhs(b0, y, k1);
#pragma unroll
    for (int c = 0; c < 5; ++c) yt[c] = y[c] + h * (A21 * k1[c]);
    rhs(b1, yt, k2);
#pragma unroll
    for (int c = 0; c < 5; ++c)
      yt[c] = y[c] + h * (A31 * k1[c] + A32 * k2[c]);
    rhs(b2, yt, k3);
#pragma unroll
    for (int c = 0; c < 5; ++c)
      yt[c] = y[c] + h * (A41 * k1[c] + A42 * k2[c] + A43 * k3[c]);
    rhs(b3, yt, k4);
#pragma unroll
    for (int c = 0; c < 5; ++c)
      yt[c] = y[c] + h * (A51 * k1[c] + A52 * k2[c] + A53 * k3[c] + A54 * k4[c]);
    rhs(b4, yt, k5);
#pragma unroll
    for (int c = 0; c < 5; ++c)
      yt[c] = y[c] + h * (A61 * k1[c] + A62 * k2[c] + A63 * k3[c] +
                          A64 * k4[c] + A65 * k5[c]);
    rhs(b5, yt, k6);
#pragma unroll
    for (int c = 0; c < 5; ++c)
      y[c] = y[c] + h * (B1 * k1[c] + B2 * k2[c] + B3 * k3[c] +
                         B4 * k4[c] + B5 * k5[c] + B6 * k6[c]);

    if ((s & 7) == 7) {
      float* o = out + (size_t)(q + 1) * 5;
#pragma unroll
      for (int c = 0; c < 5; ++c) o[c] = y[c] * sc[c];
    }
  }
}

// ---------------------------------------------------------------------------
// d_in order: 0:y0_ignored(5) 1:ts(T) 2:state_vec(5) 3:w_in(64) 4:b_in(64)
//             5:w_h(4096) 6:b_h(64) 7:w_out(64) 8:b_out(1) 9:scales(5)
// d_out: [T,5] f32.  d_ws: betas buffer (6*8*(T-1) floats ~= 192 KB).
// ---------------------------------------------------------------------------
extern "C" void kernel_launch(void* const* d_in, const int* in_sizes, int n_in,
                              void* d_out, int out_size, void* d_ws, size_t ws_size,
                              hipStream_t stream) {
  const float* ts     = (const float*)d_in[1];
  const float* state  = (const float*)d_in[2];
  const float* w_in   = (const float*)d_in[3];
  const float* b_in   = (const float*)d_in[4];
  const float* w_h    = (const float*)d_in[5];
  const float* b_h    = (const float*)d_in[6];
  const float* w_out  = (const float*)d_in[7];
  const float* b_out  = (const float*)d_in[8];
  const float* scales = (const float*)d_in[9];

  const int T    = in_sizes[1];        // 1024
  const int nsub = 8 * (T - 1);        // 8184 substeps
  const int ncol = 6 * nsub;           // 49104 beta evaluations

  float* betas = (float*)d_ws;
  float* out   = (float*)d_out;

  const int blocks = (ncol + 15) / 16; // one wave per 16 columns
  beta_mlp_wmma<<<blocks, 32, 0, stream>>>(ts, w_in, b_in, w_h, b_h,
                                           w_out, b_out, betas, ncol);
  tsit5_scan<<<1, 32, 0, stream>>>(ts, state, scales, betas, out, nsub);
}